// ECRGS_45200235823342
// MI455X (gfx1250) — compile-verified
//
#include <hip/hip_runtime.h>

// GCN encoder ×3 graphs on MI455X (gfx1250, wave32).
// Dense transforms: V_WMMA_F32_16X16X4_F32 (exact fp32 matrix path), 2 M-tiles
// per wave for B-fragment reuse. Aggregation: float4 gathers + coalesced
// global_atomic_add_f32 (working set is L2-resident: 192MB L2 vs ~51MB/layer).

typedef __attribute__((ext_vector_type(2))) float v2f;
typedef __attribute__((ext_vector_type(8))) float v8f;

#define FIN  128
#define FHID 128
#define FOUT 64

// ---------------- WMMA GEMM:  Out[row, :] = (A[row, :] @ W) * normRow[row] ----------------
// One wave per (MT x 1) group of 16x16 output tiles; B fragment loaded once per
// k-step and reused across MT WMMAs. blockDim.x = (NCOL/16)*32, grid = Nrows/(16*MT).
template <int K, int NCOL, int MT>
__global__ __launch_bounds__(256) void gemm_norm_wmma(
    const float* __restrict__ A,        // [Nrows x K] row-major
    const float* __restrict__ W,        // [K x NCOL]  row-major
    const float* __restrict__ normRow,  // [Nrows]
    float* __restrict__ Out)            // [Nrows x NCOL]
{
    const int lane = threadIdx.x & 31;
    const int wave = threadIdx.x >> 5;      // column tile
    const int l16  = lane & 15;
    const int half = lane >> 4;             // 0 or 1 (selects K pair / M+8)

    const int row0 = blockIdx.x * 16 * MT;
    const int col0 = wave * 16;

    // A 16x4 f32 layout: lane l<16 -> M=l, VGPR0/1 hold K = 2*half / 2*half+1
    // B 4x16 f32 layout (mirror):  lane l<16 -> N=l, VGPRs hold K = 2*half / 2*half+1
    const float* Wcol = W + col0 + l16;

    v8f c[MT];
#pragma unroll
    for (int m = 0; m < MT; ++m) c[m] = (v8f){};

#pragma unroll
    for (int k0 = 0; k0 < K; k0 += 4) {
        const int ka = k0 + 2 * half;
        v2f b;
        b.x = Wcol[(size_t)ka * NCOL];
        b.y = Wcol[(size_t)(ka + 1) * NCOL];
#pragma unroll
        for (int m = 0; m < MT; ++m) {
            const float* Arow = A + (size_t)(row0 + 16 * m + l16) * K;
            v2f a;
            a.x = Arow[ka];
            a.y = Arow[ka + 1];
            // 8 args: (neg_a, A, neg_b, B, c_mod, C, reuse_a, reuse_b)
            c[m] = __builtin_amdgcn_wmma_f32_16x16x4_f32(
                       false, a, false, b, (short)0, c[m], false, false);
        }
    }

    // C/D 16x16 f32 layout: VGPR r, half h -> row = base + r + 8*h, col = col0 + l16
#pragma unroll
    for (int m = 0; m < MT; ++m) {
#pragma unroll
        for (int r = 0; r < 8; ++r) {
            const int row = row0 + 16 * m + r + 8 * half;
            Out[(size_t)row * NCOL + col0 + l16] = c[m][r] * normRow[row];
        }
    }
}

// ---------------- degree: deg_src[src[e]] += 1, deg_dst[dst[e]] += 1 ----------------
__global__ __launch_bounds__(256) void degree_kernel(
    const int* __restrict__ src, const int* __restrict__ dst,
    float* __restrict__ degS, float* __restrict__ degD, int E)
{
    int e = blockIdx.x * blockDim.x + threadIdx.x;
    if (e < E) {
        atomicAdd(&degS[src[e]], 1.0f);
        atomicAdd(&degD[dst[e]], 1.0f);
    }
}

// ---------------- norm: deg -> rsqrt(max(deg,1)) in place (both arrays) ----------------
__global__ __launch_bounds__(256) void norm_kernel(
    float* __restrict__ degS, float* __restrict__ degD, int N)
{
    int n = blockIdx.x * blockDim.x + threadIdx.x;
    if (n < N) {
        degS[n] = rsqrtf(fmaxf(degS[n], 1.0f));
        degD[n] = rsqrtf(fmaxf(degD[n], 1.0f));
    }
}

// ---------------- SpMM scatter: Agg[dst[e], f] += H[src[e], f], float4 per thread ----------------
// Thread covers 4 consecutive features of one edge: b128 gather, 4 coalesced f32 atomics,
// index loads amortized 4x. totalQuads = E * (F/4).
template <int LOGF>
__global__ __launch_bounds__(256) void spmm_scatter4(
    const float* __restrict__ H, const int* __restrict__ src,
    const int* __restrict__ dst, float* __restrict__ Agg, int totalQuads)
{
    int tid = blockIdx.x * blockDim.x + threadIdx.x;
    if (tid >= totalQuads) return;
    const int e = tid >> (LOGF - 2);
    const int q = tid & ((1 << (LOGF - 2)) - 1);
    const int s = src[e];
    const int d = dst[e];
    const float4 v = *(const float4*)(H + ((size_t)s << LOGF) + (q << 2));
    float* ap = Agg + ((size_t)d << LOGF) + (q << 2);
    atomicAdd(ap + 0, v.x);
    atomicAdd(ap + 1, v.y);
    atomicAdd(ap + 2, v.z);
    atomicAdd(ap + 3, v.w);
}

// ---------------- post layer1: out = relu(agg * norm_dst[n] + bias[f]), float4 ----------------
template <int LOGF>
__global__ __launch_bounds__(256) void post_relu4(
    const float* __restrict__ Agg, const float* __restrict__ normD,
    const float* __restrict__ bias, float* __restrict__ Out, int totalQuads)
{
    int tid = blockIdx.x * blockDim.x + threadIdx.x;
    if (tid >= totalQuads) return;
    const int n = tid >> (LOGF - 2);
    const int q = tid & ((1 << (LOGF - 2)) - 1);
    const float nd = normD[n];
    const float4 a = *(const float4*)(Agg + ((size_t)tid << 2));
    const float4 b = *(const float4*)(bias + (q << 2));
    float4 o;
    o.x = fmaxf(fmaf(a.x, nd, b.x), 0.0f);
    o.y = fmaxf(fmaf(a.y, nd, b.y), 0.0f);
    o.z = fmaxf(fmaf(a.z, nd, b.z), 0.0f);
    o.w = fmaxf(fmaf(a.w, nd, b.w), 0.0f);
    *(float4*)(Out + ((size_t)tid << 2)) = o;
}

// ---------------- post layer2 (in-place on d_out slice): out = out*norm_dst + bias, float4 ----------------
template <int LOGF>
__global__ __launch_bounds__(256) void post_out4(
    float* __restrict__ OutSeg, const float* __restrict__ normD,
    const float* __restrict__ bias, int totalQuads)
{
    int tid = blockIdx.x * blockDim.x + threadIdx.x;
    if (tid >= totalQuads) return;
    const int n = tid >> (LOGF - 2);
    const int q = tid & ((1 << (LOGF - 2)) - 1);
    const float nd = normD[n];
    float4 a = *(const float4*)(OutSeg + ((size_t)tid << 2));
    const float4 b = *(const float4*)(bias + (q << 2));
    a.x = fmaf(a.x, nd, b.x);
    a.y = fmaf(a.y, nd, b.y);
    a.z = fmaf(a.z, nd, b.z);
    a.w = fmaf(a.w, nd, b.w);
    *(float4*)(OutSeg + ((size_t)tid << 2)) = a;
}

extern "C" void kernel_launch(void* const* d_in, const int* in_sizes, int n_in,
                              void* d_out, int out_size, void* d_ws, size_t ws_size,
                              hipStream_t stream) {
    // inputs: feat1,src1,dst1, feat2,src2,dst2, feat,src,dst, W1,b1,W2,b2
    const int N = in_sizes[0] / FIN;   // 100000
    const int E = in_sizes[1];         // 1600000

    const float* W1 = (const float*)d_in[9];
    const float* b1 = (const float*)d_in[10];
    const float* W2 = (const float*)d_in[11];
    const float* b2 = (const float*)d_in[12];

    // workspace layout (floats): degS[N] | degD[N] | bufA[N*FHID] | bufB[N*FHID]
    float* degS = (float*)d_ws;
    float* degD = degS + N;
    float* bufA = degD + N;
    float* bufB = bufA + (size_t)N * FHID;

    const int MT     = 2;                               // M-tiles per wave (N/(16*2)=3125 exact)
    const int mtiles = N / (16 * MT);
    const int nBlk   = (N + 255) / 256;
    const int eBlk   = (E + 255) / 256;

    for (int g = 0; g < 3; ++g) {
        const float* feat = (const float*)d_in[3 * g + 0];
        const int*   src  = (const int*)  d_in[3 * g + 1];
        const int*   dst  = (const int*)  d_in[3 * g + 2];
        float* out_g = (float*)d_out + (size_t)g * N * FOUT;

        // zero accumulators (layer-2 accumulator lives directly in the output slice)
        hipMemsetAsync(degS, 0, (size_t)N * sizeof(float), stream);
        hipMemsetAsync(degD, 0, (size_t)N * sizeof(float), stream);
        hipMemsetAsync(bufB, 0, (size_t)N * FHID * sizeof(float), stream);
        hipMemsetAsync(out_g, 0, (size_t)N * FOUT * sizeof(float), stream);

        // degrees and symmetric norms
        degree_kernel<<<eBlk, 256, 0, stream>>>(src, dst, degS, degD, E);
        norm_kernel<<<nBlk, 256, 0, stream>>>(degS, degD, N);

        // ---- layer 1: h = (X @ W1) * norm_src ; agg = scatter ; a = relu(agg*norm_dst + b1)
        gemm_norm_wmma<FIN, FHID, MT><<<mtiles, (FHID / 16) * 32, 0, stream>>>(feat, W1, degS, bufA);
        {
            const int quads = E * (FHID / 4);          // 51.2M threads
            spmm_scatter4<7><<<(quads + 255) / 256, 256, 0, stream>>>(bufA, src, dst, bufB, quads);
        }
        {
            const int quads = N * (FHID / 4);
            post_relu4<7><<<(quads + 255) / 256, 256, 0, stream>>>(bufB, degD, b1, bufA, quads);
        }

        // ---- layer 2: t = (a @ W2) * norm_src ; agg2(out) = scatter ; out = out*norm_dst + b2
        gemm_norm_wmma<FHID, FOUT, MT><<<mtiles, (FOUT / 16) * 32, 0, stream>>>(bufA, W2, degS, bufB);
        {
            const int quads = E * (FOUT / 4);          // 25.6M threads
            spmm_scatter4<6><<<(quads + 255) / 256, 256, 0, stream>>>(bufB, src, dst, out_g, quads);
        }
        {
            const int quads = N * (FOUT / 4);
            post_out4<6><<<(quads + 255) / 256, 256, 0, stream>>>(out_g, degD, b2, quads);
        }
    }
}